// Quantize_66614942761259
// MI455X (gfx1250) — compile-verified
//
#include <hip/hip_runtime.h>

typedef __attribute__((ext_vector_type(2))) float v2f;
typedef __attribute__((ext_vector_type(8))) float v8f;

#define DIM   64
#define NEMB  512
#define EMB_PITCH 66            // padded LDS pitch: banks (66n+k)%64 = (2n+k)%64
#define WAVES_PER_BLOCK 8
#define ROWS_PER_WAVE   32      // two 16-row M-tiles per wave
#define ROWS_PER_BLOCK  (WAVES_PER_BLOCK * ROWS_PER_WAVE)   // 256
#define BLOCK_THREADS   (WAVES_PER_BLOCK * 32)              // 256

// LDS layout in floats:
#define SM_E2    (NEMB * EMB_PITCH)                         // embT: [0, 33792)
#define SM_IDX   (SM_E2 + NEMB)                             // e2:   [33792, 34304)
#define SM_RED   (SM_IDX + ROWS_PER_BLOCK)                  // widx (int): 256
#define SM_TOTAL (SM_RED + BLOCK_THREADS)                   // red: 256 floats

__global__ void __launch_bounds__(BLOCK_THREADS)
vq_wmma_kernel(const float* __restrict__ x,
               const float* __restrict__ embed,   // [DIM][NEMB] row-major
               float* __restrict__ out_q,         // [N][DIM]
               float* __restrict__ out_ind,       // [N] (indices as float)
               float* __restrict__ partial)       // [gridDim.x] block partial SSE
{
    extern __shared__ float smem[];
    float* embT = smem;                       // embT[n*EMB_PITCH + k] = embed[k][n]
    float* e2s  = smem + SM_E2;               // ||e_n||^2
    int*   widx = (int*)(smem + SM_IDX);      // per-row argmin
    float* red  = smem + SM_RED;              // block reduction scratch

    const int tid = threadIdx.x;

    // ---- Stage codebook into LDS, transposed (coalesced global reads) ----
    for (int i = tid; i < DIM * NEMB; i += BLOCK_THREADS) {
        const int k = i >> 9;              // NEMB == 512
        const int n = i & (NEMB - 1);
        embT[n * EMB_PITCH + k] = embed[i];
    }
    __syncthreads();

    // ---- ||e_n||^2 ----
    for (int n = tid; n < NEMB; n += BLOCK_THREADS) {
        float s = 0.f;
        #pragma unroll
        for (int k = 0; k < DIM; ++k) {
            const float v = embT[n * EMB_PITCH + k];
            s = fmaf(v, v, s);
        }
        e2s[n] = s;
    }
    __syncthreads();

    const int wave = tid >> 5;
    const int lane = tid & 31;
    const int half = lane >> 4;     // which 16-lane half
    const int m    = lane & 15;     // row (A) / column (B,D) within tile
    const long rowbase = (long)blockIdx.x * ROWS_PER_BLOCK + (long)wave * ROWS_PER_WAVE;

    // ---- A fragments for two 16x64 fp32 M-tiles (ISA 16x4 layout:
    //      lane half h, VGPR j -> K = 4c + 2h + j) ----
    v2f afrag0[16], afrag1[16];
    {
        const float* xr0 = x + (rowbase + m) * DIM + 2 * half;
        const float* xr1 = x + (rowbase + 16 + m) * DIM + 2 * half;
        #pragma unroll
        for (int c = 0; c < 16; ++c) {
            afrag0[c] = *(const v2f*)(xr0 + 4 * c);
            afrag1[c] = *(const v2f*)(xr1 + 4 * c);
        }
    }

    float minv0[8], minv1[8];
    int   mini0[8], mini1[8];
    #pragma unroll
    for (int v = 0; v < 8; ++v) {
        minv0[v] = 3.4e38f; mini0[v] = 0;
        minv1[v] = 3.4e38f; mini1[v] = 0;
    }

    // ---- Sweep 32 column tiles; each B fragment feeds 2 WMMAs (2 M-tiles),
    //      4 independent accumulator chains hide DS + WMMA latency ----
    for (int t = 0; t < NEMB / 16; ++t) {
        v8f acc00 = {0.f,0.f,0.f,0.f,0.f,0.f,0.f,0.f};
        v8f acc01 = {0.f,0.f,0.f,0.f,0.f,0.f,0.f,0.f};
        v8f acc10 = {0.f,0.f,0.f,0.f,0.f,0.f,0.f,0.f};
        v8f acc11 = {0.f,0.f,0.f,0.f,0.f,0.f,0.f,0.f};
        const float* bcol = embT + (t * 16 + m) * EMB_PITCH + 2 * half;
        #pragma unroll
        for (int c = 0; c < 16; c += 2) {
            v2f b0, b1;
            b0[0] = bcol[4 * c + 0];
            b0[1] = bcol[4 * c + 1];
            b1[0] = bcol[4 * c + 4];
            b1[1] = bcol[4 * c + 5];
            acc00 = __builtin_amdgcn_wmma_f32_16x16x4_f32(
                false, afrag0[c + 0], false, b0, (short)0, acc00, false, false);
            acc10 = __builtin_amdgcn_wmma_f32_16x16x4_f32(
                false, afrag1[c + 0], false, b0, (short)0, acc10, false, false);
            acc01 = __builtin_amdgcn_wmma_f32_16x16x4_f32(
                false, afrag0[c + 1], false, b1, (short)0, acc01, false, false);
            acc11 = __builtin_amdgcn_wmma_f32_16x16x4_f32(
                false, afrag1[c + 1], false, b1, (short)0, acc11, false, false);
        }
        // D element (vgpr v, lane) = (M = v + 8*half, N = t*16 + m)
        const float e2v = e2s[t * 16 + m];
        const int   nid = t * 16 + m;
        #pragma unroll
        for (int v = 0; v < 8; ++v) {
            const float d0 = fmaf(-2.f, acc00[v] + acc01[v], e2v);
            if (d0 < minv0[v]) { minv0[v] = d0; mini0[v] = nid; }
            const float d1 = fmaf(-2.f, acc10[v] + acc11[v], e2v);
            if (d1 < minv1[v]) { minv1[v] = d1; mini1[v] = nid; }
        }
    }

    // ---- Argmin across the 16 lanes of each half (index tie-break: lowest) ----
    #pragma unroll
    for (int off = 1; off < 16; off <<= 1) {
        #pragma unroll
        for (int v = 0; v < 8; ++v) {
            float ov; int oi;
            ov = __shfl_xor(minv0[v], off, 32);
            oi = __shfl_xor(mini0[v], off, 32);
            if (ov < minv0[v] || (ov == minv0[v] && oi < mini0[v])) {
                minv0[v] = ov; mini0[v] = oi;
            }
            ov = __shfl_xor(minv1[v], off, 32);
            oi = __shfl_xor(mini1[v], off, 32);
            if (ov < minv1[v] || (ov == minv1[v] && oi < mini1[v])) {
                minv1[v] = ov; mini1[v] = oi;
            }
        }
    }
    if (m == 0) {
        #pragma unroll
        for (int v = 0; v < 8; ++v) {
            widx[wave * ROWS_PER_WAVE + half * 8 + v]      = mini0[v];
            widx[wave * ROWS_PER_WAVE + 16 + half * 8 + v] = mini1[v];
        }
    }
    __syncthreads();

    // ---- Gather quantize rows from LDS, write outputs, accumulate SSE ----
    float ss = 0.f;
    #pragma unroll
    for (int g = 0; g < 2; ++g) {
        const int r    = g * 16 + (lane >> 1);   // row within wave tile-pair
        const int part = lane & 1;               // column half (0..31 / 32..63)
        const long row = rowbase + r;
        const int idx  = widx[wave * ROWS_PER_WAVE + r];
        const float* qsrc = embT + idx * EMB_PITCH + part * 32;
        const float* xsrc = x + row * DIM + part * 32;
        float*       qdst = out_q + row * DIM + part * 32;
        #pragma unroll
        for (int j = 0; j < 32; ++j) {
            const float q  = qsrc[j];
            const float xv = xsrc[j];
            qdst[j] = q;
            const float dlt = q - xv;
            ss = fmaf(dlt, dlt, ss);
        }
        if (part == 0) out_ind[row] = (float)idx;
    }

    // ---- Deterministic block reduction of SSE -> partial[block] ----
    red[tid] = ss;
    __syncthreads();
    for (int s2 = BLOCK_THREADS / 2; s2 > 0; s2 >>= 1) {
        if (tid < s2) red[tid] += red[tid + s2];
        __syncthreads();
    }
    if (tid == 0) partial[blockIdx.x] = red[0];
}

__global__ void vq_diff_reduce(const float* __restrict__ partial, int nparts,
                               float* __restrict__ out_diff, float inv_count)
{
    __shared__ float red[256];
    float s = 0.f;
    for (int i = threadIdx.x; i < nparts; i += 256) s += partial[i];
    red[threadIdx.x] = s;
    __syncthreads();
    for (int k = 128; k > 0; k >>= 1) {
        if (threadIdx.x < k) red[threadIdx.x] += red[threadIdx.x + k];
        __syncthreads();
    }
    if (threadIdx.x == 0) *out_diff = red[0] * inv_count;
}

extern "C" void kernel_launch(void* const* d_in, const int* in_sizes, int n_in,
                              void* d_out, int out_size, void* d_ws, size_t ws_size,
                              hipStream_t stream) {
    const float* x     = (const float*)d_in[0];   // [N, DIM] flattened
    const float* embed = (const float*)d_in[1];   // [DIM, NEMB]
    const int N = in_sizes[0] / DIM;              // 131072

    float* out      = (float*)d_out;
    float* out_q    = out;                        // N*DIM floats
    float* out_diff = out + (size_t)N * DIM;      // 1 float
    float* out_ind  = out + (size_t)N * DIM + 1;  // N floats (indices)

    float* partial = (float*)d_ws;                // gridDim partial sums

    const int blocks = N / ROWS_PER_BLOCK;        // 512
    const size_t shmem = (size_t)SM_TOTAL * sizeof(float);  // ~136 KB

    vq_wmma_kernel<<<blocks, BLOCK_THREADS, shmem, stream>>>(
        x, embed, out_q, out_ind, partial);
    vq_diff_reduce<<<1, 256, 0, stream>>>(
        partial, blocks, out_diff, 1.0f / (float)((size_t)N * DIM));
}